// ForgetMult_84774064488815
// MI455X (gfx1250) — compile-verified
//
#include <hip/hip_runtime.h>
#include <hip/hip_bf16.h>
#include <stdint.h>

// Problem geometry (fixed by the reference)
#define S_DIM   2048
#define N_COLS  16384            // B*H = 16*1024 channels
#define CHUNKS  64               // time chunks
#define L_STEPS 32               // S_DIM / CHUNKS
#define CG      (N_COLS / 4)     // 4-channel groups = 4096
#define PD      8                // prefetch distance (time steps)
#define ADEPTH  4                // async global->LDS pipeline depth (time steps)

typedef float v4f __attribute__((ext_vector_type(4)));

__device__ __forceinline__ float4 ld4(const float* p) {
    return *reinterpret_cast<const float4*>(p);
}
__device__ __forceinline__ void st4(float* p, float4 v) {
    *reinterpret_cast<float4*>(p) = v;
}

// ---- CDNA5 async global->LDS (ASYNCcnt-tracked) helpers ---------------------
// Each lane copies its own 16 B; VDST operand is the LDS byte address VGPR.
__device__ __forceinline__ void async_load16(unsigned lds_byte_addr, const float* gptr) {
    asm volatile("global_load_async_to_lds_b128 %0, %1, off"
                 :: "v"(lds_byte_addr),
                    "v"((unsigned long long)(uintptr_t)gptr)
                 : "memory");
}
// Wait until at most N async ops remain outstanding (template => guaranteed
// immediate for the SOPP encoding).  Async loads complete in order, so
// ASYNCcnt <= N proves the oldest (issued-first) ops have landed in LDS.
template <int N>
__device__ __forceinline__ void wait_async_le() {
    asm volatile("s_wait_asynccnt %0" :: "n"(N) : "memory");
}
// Ensure our ds_load readback finished before the async engine reuses the slot.
__device__ __forceinline__ void wait_ds0() {
    asm volatile("s_wait_dscnt 0x0" ::: "memory");
}

// ---------------------------------------------------------------------------
// Pass 1: per (chunk, 4-channel group), compute chunk-aggregate affine map
//   h_end = A * h_start + B   with per-step a = 1-f, b = f*x
// Streaming is staged through LDS with a 4-deep async ring buffer; each thread
// only ever touches its own LDS slot, so no workgroup barriers are needed.
// Chunk order ascends with blockIdx => L2 ends up holding the high-chunk tail.
// ---------------------------------------------------------------------------
__global__ __launch_bounds__(256) void fm_pass1(const float* __restrict__ f,
                                                const float* __restrict__ x,
                                                float* __restrict__ aggA,
                                                float* __restrict__ aggB) {
    __shared__ __align__(16) float sf[ADEPTH * 1024];   // 16 KB
    __shared__ __align__(16) float sx[ADEPTH * 1024];   // 16 KB

    const int tid = blockIdx.x * blockDim.x + threadIdx.x;   // 0 .. CG*CHUNKS-1
    const int cg  = tid & (CG - 1);                          // channel group (fast -> coalesced)
    const int c   = tid >> 12;                               // chunk id (ascending with blockIdx)
    const int j0  = cg << 2;                                 // first channel of group
    const size_t base = (size_t)c * L_STEPS * N_COLS + (size_t)j0;

    const int lidx = threadIdx.x << 2;                       // my 4 floats in a slot
    const unsigned lf = (unsigned)(unsigned long long)(uintptr_t)&sf[lidx];
    const unsigned lx = (unsigned)(unsigned long long)(uintptr_t)&sx[lidx];

    // Prologue: fill the ring (2 async ops per step, ADEPTH steps => 8 in flight)
#pragma unroll
    for (int d = 0; d < ADEPTH; ++d) {
        async_load16(lf + (unsigned)d * 4096u, f + base + (size_t)d * N_COLS);
        async_load16(lx + (unsigned)d * 4096u, x + base + (size_t)d * N_COLS);
    }

    float4 A  = make_float4(1.f, 1.f, 1.f, 1.f);
    float4 Bc = make_float4(0.f, 0.f, 0.f, 0.f);

#pragma unroll
    for (int t = 0; t < L_STEPS; ++t) {
        // Outstanding pairs cover slots t .. min(t+ADEPTH-1, L-1).  To prove
        // slot t's (f,x) pair landed, allow only the *younger* pairs to remain:
        //   allowed = 2 * min(ADEPTH-1, L-1-t)   ->  6,...,6,4,2,0
        constexpr int kSteady = 2 * (ADEPTH - 1);
        const int remain = 2 * (L_STEPS - 1 - t);
        if (remain >= kSteady)      wait_async_le<kSteady>();
        else if (remain == 4)       wait_async_le<4>();
        else if (remain == 2)       wait_async_le<2>();
        else                        wait_async_le<0>();

        const int slot = t & (ADEPTH - 1);
        const float4 fv = ld4(&sf[slot * 1024 + lidx]);      // ds_load_b128
        const float4 xv = ld4(&sx[slot * 1024 + lidx]);
#define P1(c_) { const float a_ = 1.0f - fv.c_;                    \
                 A.c_  *= a_;                                      \
                 Bc.c_  = fmaf(Bc.c_, a_, fv.c_ * xv.c_); }
        P1(x) P1(y) P1(z) P1(w)
#undef P1
        if (t + ADEPTH < L_STEPS) {
            wait_ds0();                                      // slot readback done
            async_load16(lf + (unsigned)slot * 4096u,
                         f + base + (size_t)(t + ADEPTH) * N_COLS);
            async_load16(lx + (unsigned)slot * 4096u,
                         x + base + (size_t)(t + ADEPTH) * N_COLS);
        }
    }

    st4(aggA + (size_t)c * N_COLS + j0, A);
    st4(aggB + (size_t)c * N_COLS + j0, Bc);
}

// ---------------------------------------------------------------------------
// Pass 2: sequentially compose the 64 chunk aggregates per channel to get the
// hidden state entering each chunk.  4096 threads; aggregates are L2-hot.
// ---------------------------------------------------------------------------
__global__ __launch_bounds__(256) void fm_pass2(const float* __restrict__ h0,
                                                const float* __restrict__ aggA,
                                                const float* __restrict__ aggB,
                                                float* __restrict__ hstart) {
    const int tid = blockIdx.x * blockDim.x + threadIdx.x;   // 0 .. CG-1
    const int j0  = tid << 2;

    float4 h = ld4(h0 + j0);
#pragma unroll 4
    for (int c = 0; c < CHUNKS; ++c) {
        st4(hstart + (size_t)c * N_COLS + j0, h);
        const float4 Av = ld4(aggA + (size_t)c * N_COLS + j0);
        const float4 Bv = ld4(aggB + (size_t)c * N_COLS + j0);
#define P2(c_) h.c_ = fmaf(Av.c_, h.c_, Bv.c_);
        P2(x) P2(y) P2(z) P2(w)
#undef P2
    }
}

// ---------------------------------------------------------------------------
// Pass 3: re-scan each chunk from its known start state and emit every h_t.
//   h = f*(x - h) + h   ==  f*x + (1-f)*h
// Chunk order is REVERSED (63 -> 0): the first blocks consume the f,x tail
// that pass 1 left resident in the 192 MB L2; pass 3 in turn ends on the low
// chunks, which the next replay's pass 1 reads first.  The write-once output
// is stored non-temporally so it doesn't evict the reused f,x lines.
// ---------------------------------------------------------------------------
__global__ __launch_bounds__(256) void fm_pass3(const float* __restrict__ f,
                                                const float* __restrict__ x,
                                                const float* __restrict__ hstart,
                                                float* __restrict__ out) {
    const int tid = blockIdx.x * blockDim.x + threadIdx.x;
    const int cg  = tid & (CG - 1);
    const int c   = (CHUNKS - 1) - (tid >> 12);              // reversed chunk order
    const int j0  = cg << 2;
    const size_t base = (size_t)c * L_STEPS * N_COLS + (size_t)j0;

    v4f h = *reinterpret_cast<const v4f*>(hstart + (size_t)c * N_COLS + j0);

#pragma unroll
    for (int t = 0; t < L_STEPS; ++t) {
        const size_t off = base + (size_t)t * N_COLS;
        if (t + PD < L_STEPS) {
            __builtin_prefetch(f + base + (size_t)(t + PD) * N_COLS, 0, 1);
            __builtin_prefetch(x + base + (size_t)(t + PD) * N_COLS, 0, 1);
        }
        const v4f fv = *reinterpret_cast<const v4f*>(f + off);
        const v4f xv = *reinterpret_cast<const v4f*>(x + off);
        h = fv * (xv - h) + h;                               // elementwise FMA chain
        __builtin_nontemporal_store(h, reinterpret_cast<v4f*>(out + off));
    }
}

// ---------------------------------------------------------------------------
extern "C" void kernel_launch(void* const* d_in, const int* in_sizes, int n_in,
                              void* d_out, int out_size, void* d_ws, size_t ws_size,
                              hipStream_t stream) {
    (void)in_sizes; (void)n_in; (void)out_size; (void)ws_size;

    const float* f  = (const float*)d_in[0];   // [S,B,H] fp32
    const float* x  = (const float*)d_in[1];   // [S,B,H] fp32
    const float* h0 = (const float*)d_in[2];   // [B,H]   fp32
    float* out = (float*)d_out;                // [S,B,H] fp32

    // Workspace layout: aggA | aggB | hstart, each CHUNKS*N_COLS floats (4 MB)
    float* aggA   = (float*)d_ws;
    float* aggB   = aggA + (size_t)CHUNKS * N_COLS;
    float* hstart = aggB + (size_t)CHUNKS * N_COLS;

    const dim3 blk(256);
    const dim3 grid13((CG * CHUNKS) / 256);    // 1024 blocks
    const dim3 grid2(CG / 256);                // 16 blocks

    fm_pass1<<<grid13, blk, 0, stream>>>(f, x, aggA, aggB);
    fm_pass2<<<grid2,  blk, 0, stream>>>(h0, aggA, aggB, hstart);
    fm_pass3<<<grid13, blk, 0, stream>>>(f, x, hstart, out);
}